// DynamicStateLoss_22634477650528
// MI455X (gfx1250) — compile-verified
//
#include <hip/hip_runtime.h>
#include <cstdint>

#define N_COLS 8192
#define TPB 256
#define NWAVES (TPB / 32)
#define CHUNKS (N_COLS / (TPB * 4)) // 8 float4 chunks per thread per row
#define RPB 2                       // rows per block (double-buffered pipeline)

__global__ __launch_bounds__(64) void dsl_init(double* tot, unsigned* cnt) {
    if (threadIdx.x == 0) { *tot = 0.0; *cnt = 0u; }
}

__device__ __forceinline__ void acc_one(float s, int m, unsigned char v,
                                        float& tmax, float& tpos, int& tcnt) {
    const bool vb = (v != 0);
    const bool pb = vb && (m > 0);
    tmax = fmaxf(tmax, vb ? s : -__builtin_inff());
    tcnt += pb ? 1 : 0;
    tpos += pb ? s : 0.0f;
}

__global__ __launch_bounds__(TPB) void dsl_rows(
    const float* __restrict__ scores,
    const int* __restrict__ som,
    const unsigned char* __restrict__ valid,
    double* __restrict__ tot, unsigned* __restrict__ cnt)
{
    __shared__ float sS[RPB][N_COLS];   // 2 x 32 KB double-buffered score stage
    __shared__ float wmax[NWAVES], wpos[NWAVES], wsum[NWAVES];
    __shared__ int   wcnt[NWAVES];

    const int tid = threadIdx.x;
    const size_t base = (size_t)blockIdx.x * RPB * (size_t)N_COLS;

    // ---- Issue ALL async global->LDS copies up front (per-wave, in-order) ----
#pragma unroll
    for (int r = 0; r < RPB; ++r) {
#pragma unroll
        for (int i = 0; i < CHUNKS; ++i) {
            const int e = (tid + i * TPB) * 4;
            const float* gp = scores + base + (size_t)r * N_COLS + e;
            const uint32_t laddr = (uint32_t)(uintptr_t)(&sS[r][e]);
            asm volatile("global_load_async_to_lds_b128 %0, %1, off"
                         :: "v"(laddr), "v"(gp) : "memory");
        }
    }

    // ---- Masks for both rows -> registers, in flight behind the copies ----
    int4   mv[RPB][CHUNKS];
    uchar4 vv[RPB][CHUNKS];
#pragma unroll
    for (int r = 0; r < RPB; ++r) {
#pragma unroll
        for (int i = 0; i < CHUNKS; ++i) {
            const int e = (tid + i * TPB) * 4;
            mv[r][i] = *(const int4*)(som + base + (size_t)r * N_COLS + e);
            vv[r][i] = *(const uchar4*)(valid + base + (size_t)r * N_COLS + e);
        }
    }

    double   block_total = 0.0;
    unsigned block_count = 0u;

#pragma unroll
    for (int r = 0; r < RPB; ++r) {
        // Threshold wait: row 0 only needs ASYNCcnt <= 8 (its chunks landed,
        // row 1's copies keep streaming behind row 0's compute).
        if (r == 0) asm volatile("s_wait_asynccnt 0x8" ::: "memory");
        else        asm volatile("s_wait_asynccnt 0x0" ::: "memory");

        // ---- Pass A: valid-masked max, positive count, positive score sum ----
        float tmax = -__builtin_inff();
        float tpos = 0.0f;
        int   tcnt = 0;
#pragma unroll
        for (int i = 0; i < CHUNKS; ++i) {
            const int e = (tid + i * TPB) * 4;
            const float4 s = *(const float4*)(&sS[r][e]);
            acc_one(s.x, mv[r][i].x, vv[r][i].x, tmax, tpos, tcnt);
            acc_one(s.y, mv[r][i].y, vv[r][i].y, tmax, tpos, tcnt);
            acc_one(s.z, mv[r][i].z, vv[r][i].z, tmax, tpos, tcnt);
            acc_one(s.w, mv[r][i].w, vv[r][i].w, tmax, tpos, tcnt);
        }
        // wave32 butterfly reduction
#pragma unroll
        for (int off = 16; off > 0; off >>= 1) {
            tmax = fmaxf(tmax, __shfl_xor(tmax, off, 32));
            tpos += __shfl_xor(tpos, off, 32);
            tcnt += __shfl_xor(tcnt, off, 32);
        }
        const int wid = tid >> 5;
        if ((tid & 31) == 0) { wmax[wid] = tmax; wpos[wid] = tpos; wcnt[wid] = tcnt; }
        __syncthreads();

        float bmax = wmax[0];
#pragma unroll
        for (int k = 1; k < NWAVES; ++k) bmax = fmaxf(bmax, wmax[k]);

        // ---- Pass B: sum of exp(s - bmax) over valid ----
        float tsum = 0.0f;
#pragma unroll
        for (int i = 0; i < CHUNKS; ++i) {
            const int e = (tid + i * TPB) * 4;
            const float4 s = *(const float4*)(&sS[r][e]);
            tsum += (vv[r][i].x ? __expf(s.x - bmax) : 0.0f);
            tsum += (vv[r][i].y ? __expf(s.y - bmax) : 0.0f);
            tsum += (vv[r][i].z ? __expf(s.z - bmax) : 0.0f);
            tsum += (vv[r][i].w ? __expf(s.w - bmax) : 0.0f);
        }
#pragma unroll
        for (int off = 16; off > 0; off >>= 1) tsum += __shfl_xor(tsum, off, 32);
        if ((tid & 31) == 0) wsum[wid] = tsum;
        __syncthreads();

        if (tid == 0) {
            float bsum = 0.0f, bpos = 0.0f; int bcnt = 0;
#pragma unroll
            for (int k = 0; k < NWAVES; ++k) { bsum += wsum[k]; bpos += wpos[k]; bcnt += wcnt[k]; }
            const float lse = bmax + logf(bsum);
            block_total += (double)bcnt * (double)lse - (double)bpos;
            block_count += (unsigned)bcnt;
        }
        // Keep fast waves from overwriting wmax/wpos/wcnt for the next row
        // while thread 0 is still reading this row's reduction slots.
        __syncthreads();
    }

    if (tid == 0) {
        atomicAdd(tot, block_total);
        atomicAdd(cnt, block_count);
    }
}

__global__ __launch_bounds__(64) void dsl_final(const double* __restrict__ tot,
                                                const unsigned* __restrict__ cnt,
                                                float* __restrict__ out) {
    if (threadIdx.x == 0) {
        const unsigned c = *cnt;
        out[0] = (c > 0) ? (float)(*tot / (double)c) : 0.0f;
    }
}

extern "C" void kernel_launch(void* const* d_in, const int* in_sizes, int n_in,
                              void* d_out, int out_size, void* d_ws, size_t ws_size,
                              hipStream_t stream) {
    const float*         scores = (const float*)d_in[0];
    const int*           som    = (const int*)d_in[1];
    const unsigned char* valid  = (const unsigned char*)d_in[2]; // jnp bool = 1 byte
    float* out = (float*)d_out;

    double*   tot = (double*)d_ws;
    unsigned* cnt = (unsigned*)((char*)d_ws + sizeof(double));

    const int Brows = in_sizes[0] / N_COLS; // 4096 rows
    const int nblk  = Brows / RPB;          // 2048 blocks, 2 rows each

    dsl_init<<<1, 64, 0, stream>>>(tot, cnt);
    dsl_rows<<<nblk, TPB, 0, stream>>>(scores, som, valid, tot, cnt);
    dsl_final<<<1, 64, 0, stream>>>(tot, cnt, out);
}